// MaxPooling_47047071760693
// MI455X (gfx1250) — compile-verified
//
#include <hip/hip_runtime.h>
#include <stdint.h>

// MaxPooling over masked neighbors, N=2048 nodes, F=64 features.
// out[i,f] = max_j( inputs[i,f] * adj[i,j] )
//          = !hasOne(i) ? 0 : hasZero(i) ? max(inputs[i,f],0) : inputs[i,f]
// where hasOne/hasZero are row-wise predicates on the {0.0,1.0} adjacency.
//
// Memory-bound: ~17 MiB total traffic -> ~0.73 us at 23.3 TB/s.
// CDNA5 path: GLOBAL_LOAD_ASYNC_TO_LDS_B128 streaming (ASYNCcnt) + ds_load_b128
// reduction. The INST_OFFSET immediate is applied to both the global source and
// the LDS destination (CDNA5 ISA), so all 16 issues share one
// VGPR address pair + one LDS-offset VGPR -> zero per-issue address VALU.
// WMMA deliberately not used: no matmul-shaped work; the matrix pipe cannot
// raise the bandwidth ceiling of a streaming reduction.

#define N_NODES 2048
#define N_FEAT  64
#define WAVES_PER_BLOCK 4
#define THREADS_PER_BLOCK (WAVES_PER_BLOCK * 32)
#define ROW_BYTES (N_NODES * 4)            // 8192 B per adjacency row

// One async b128 issue moves 32 lanes * 16 B = 512 B; offset: immediate strides
// both the global and LDS addresses.
#define ASYNC_CHUNK(OFF)                                                      \
  asm volatile("global_load_async_to_lds_b128 %0, %1, off offset:" #OFF      \
               :: "v"(lds_off), "v"(gaddr) : "memory")

__global__ __launch_bounds__(THREADS_PER_BLOCK)
void maxpool_rowflags_kernel(const float* __restrict__ inputs,
                             const float* __restrict__ adj,
                             float* __restrict__ out)
{
  // Per-wave 8 KiB staging buffer (4 waves -> 32 KiB/block; WGP has 320 KiB).
  __shared__ __align__(16) unsigned char smem[WAVES_PER_BLOCK][ROW_BYTES];

  const int lane = threadIdx.x & 31;
  const int wave = threadIdx.x >> 5;
  const int row  = blockIdx.x * WAVES_PER_BLOCK + wave;   // grid covers all 2048 rows

  // ---------------- stage adj[row, :] into LDS with async b128 loads --------
  // Generic pointer to __shared__ carries the LDS byte offset in its low 32
  // bits (aperture lives in the high half).
  unsigned lds_off = (unsigned)(uintptr_t)(&smem[wave][0]) + (unsigned)(lane * 16);
  uint64_t gaddr   = (uint64_t)(uintptr_t)(adj + (size_t)row * N_NODES)
                   + (unsigned)(lane * 16);

  ASYNC_CHUNK(0);    ASYNC_CHUNK(512);  ASYNC_CHUNK(1024); ASYNC_CHUNK(1536);
  ASYNC_CHUNK(2048); ASYNC_CHUNK(2560); ASYNC_CHUNK(3072); ASYNC_CHUNK(3584);
  ASYNC_CHUNK(4096); ASYNC_CHUNK(4608); ASYNC_CHUNK(5120); ASYNC_CHUNK(5632);
  ASYNC_CHUNK(6144); ASYNC_CHUNK(6656); ASYNC_CHUNK(7168); ASYNC_CHUNK(7680);

  // Wave-private wait: all 16 async transfers into our LDS slice complete.
  asm volatile("s_wait_asynccnt 0x0" ::: "memory");

  // ---------------- OR/AND bit reduction over the row -----------------------
  // adj values are exactly 0x00000000 (0.0f) or 0x3F800000 (1.0f):
  //   OR  != 0          <=> row contains a 1.0
  //   AND != 0x3F800000 <=> row contains a 0.0
  const uint4* srow = (const uint4*)(&smem[wave][0]);
  unsigned orv = 0u, andv = 0xFFFFFFFFu;
  #pragma unroll
  for (int c = 0; c < (N_NODES / 4) / 32; ++c) {   // 16 x uint4 per lane (ds_load_b128)
    uint4 v = srow[lane + c * 32];
    orv  |= v.x | v.y | v.z | v.w;
    andv &= v.x & v.y & v.z & v.w;
  }
  // wave32 butterfly; result lands in every lane.
  #pragma unroll
  for (int m = 16; m >= 1; m >>= 1) {
    orv  |= __shfl_xor(orv,  m, 32);
    andv &= __shfl_xor(andv, m, 32);
  }
  const bool hasOne  = (orv  != 0u);
  const bool hasZero = (andv != 0x3F800000u);

  // ---------------- emit 64 features for this row (2 per lane, coalesced) ---
  const float2 x = ((const float2*)(inputs + (size_t)row * N_FEAT))[lane];
  float2 r;
  if (!hasOne) {
    r.x = 0.0f; r.y = 0.0f;                 // every masked value is 0
  } else if (hasZero) {
    r.x = fmaxf(x.x, 0.0f);                 // candidates: {x, 0}
    r.y = fmaxf(x.y, 0.0f);
  } else {
    r = x;                                  // all-ones row: max is x itself
  }
  ((float2*)(out + (size_t)row * N_FEAT))[lane] = r;
}

extern "C" void kernel_launch(void* const* d_in, const int* in_sizes, int n_in,
                              void* d_out, int out_size, void* d_ws, size_t ws_size,
                              hipStream_t stream) {
  const float* inputs = (const float*)d_in[0];   // [2048, 64]
  const float* adj    = (const float*)d_in[1];   // [2048, 2048]
  float*       out    = (float*)d_out;           // [2048, 64]

  dim3 grid(N_NODES / WAVES_PER_BLOCK);          // 512 blocks
  dim3 block(THREADS_PER_BLOCK);                 // 128 threads = 4 wave32
  hipLaunchKernelGGL(maxpool_rowflags_kernel, grid, block, 0, stream,
                     inputs, adj, out);
}